// Example3_2954937500143
// MI455X (gfx1250) — compile-verified
//
#include <hip/hip_runtime.h>
#include <math.h>

#define VOCAB 512
#define CTX   1024
#define NLAYERS 4
#define DDIM  1536
#define BATCH 32

typedef __attribute__((ext_vector_type(2))) float v2f;
typedef __attribute__((ext_vector_type(8))) float v8f;

// GEMM tiling: block = 256 threads = 8 wave32s, arranged 4 (M) x 2 (N).
// Block tile: 64 x 32 output, K staged through LDS in chunks of 32,
// double-buffered via GLOBAL_LOAD_ASYNC_TO_LDS_B128 (ASYNCcnt).
#define BM 64
#define BN 32
#define KC 32
#define LDST 36   // padded LDS row stride (floats): 144B rows -> 16B aligned,
                  // and 36*r mod 64 distinct for r=0..15 -> conflict-free frags

// ---- CDNA5 async global->LDS DMA (direct to LDS, tracked by ASYNCcnt) ----
__device__ __forceinline__ unsigned lds_off32(const void* p) {
    // Flat LDS aperture: addr[31:0] is the LDS byte offset.
    return (unsigned)(unsigned long long)p;
}

__device__ __forceinline__ void async_ld_b128(unsigned ldsaddr, const float* sbase, int voff) {
    // GVS mode: LDS address in vdst; global address is SGPR base plus VGPR offset
    asm volatile("global_load_async_to_lds_b128 %0, %1, %2"
                 :: "v"(ldsaddr), "v"(voff), "s"(sbase)
                 : "memory");
}

template<int N>
__device__ __forceinline__ void wait_asynccnt_le() {
#if __has_builtin(__builtin_amdgcn_s_wait_asynccnt)
    __builtin_amdgcn_s_wait_asynccnt((short)N);
#else
    asm volatile("s_wait_asynccnt %0" :: "i"(N) : "memory");
#endif
}

// C = A(MxK) * B(KxN)      (TRANS_B = false, B row-major KxN)
// C = A(MxK) * B(NxK)^T    (TRANS_B = true,  B row-major NxK)
// CAUSAL: fuse additive causal mask (col > row -> -1e9) into the store.
template<bool TRANS_B, bool CAUSAL>
__global__ __launch_bounds__(256)
void gemm_wmma_f32(const float* __restrict__ A, const float* __restrict__ B,
                   float* __restrict__ C,
                   int M, int Nn, int K,
                   long lda, long ldb, long ldc,
                   long strideA, long strideB, long strideC)
{
    __shared__ float As[2][BM * LDST];
    __shared__ float Bs[2][KC * LDST];   // TRANS_B: stored as [n][k] (K-major)

    const int tid  = threadIdx.x;
    const int wave = tid >> 5;
    const int lane = tid & 31;
    const int half = lane >> 4;   // 0: lanes 0-15, 1: lanes 16-31
    const int lm   = lane & 15;
    const int mt   = wave & 3;    // M tile index within block (0..3)
    const int nt   = wave >> 2;   // N tile index within block (0..1)

    const int  rowBase = blockIdx.x * BM;
    const int  colBase = blockIdx.y * BN;
    const long batch   = blockIdx.z;

    const float* Ab = A + batch * strideA;   // uniform -> SGPR base for DMA
    const float* Bb = B + batch * strideB;
    float*       Cb = C + batch * strideC;

    // Per-thread DMA coordinates (one b128 = 4 floats per lane).
    const int ar0 = tid >> 3;            // A rows 0..31 (first half)
    const int ar1 = ar0 + 32;            // A rows 32..63 (second half)
    const int ak4 = (tid & 7) << 2;      // k offset 0,4,...,28
    int agr0 = rowBase + ar0; if (agr0 >= M) agr0 = M - 1;  // clamp: keep EXEC full
    int agr1 = rowBase + ar1; if (agr1 >= M) agr1 = M - 1;
    const int br  = tid >> 3;            // B: k-row (NN) or n-col (NT), 0..31
    const int bo4 = (tid & 7) << 2;

    const unsigned asO[2] = { lds_off32(&As[0][0]), lds_off32(&As[1][0]) };
    const unsigned bsO[2] = { lds_off32(&Bs[0][0]), lds_off32(&Bs[1][0]) };

    auto stage = [&](int kt, int buf) {   // 3 async ops per wave per stage
        const int kBase = kt * KC;
        async_ld_b128(asO[buf] + (unsigned)(ar0 * LDST + ak4) * 4u, Ab,
                      (int)(((long)agr0 * lda + kBase + ak4) * 4));
        async_ld_b128(asO[buf] + (unsigned)(ar1 * LDST + ak4) * 4u, Ab,
                      (int)(((long)agr1 * lda + kBase + ak4) * 4));
        if (!TRANS_B)
            async_ld_b128(bsO[buf] + (unsigned)(br * LDST + bo4) * 4u, Bb,
                          (int)(((long)(kBase + br) * ldb + colBase + bo4) * 4));
        else
            async_ld_b128(bsO[buf] + (unsigned)(br * LDST + bo4) * 4u, Bb,
                          (int)(((long)(colBase + br) * ldb + kBase + bo4) * 4));
    };

    v8f acc = {};
    const int ktiles = K / KC;

    stage(0, 0);
    for (int kt = 0; kt < ktiles; ++kt) {
        const int buf = kt & 1;
        __syncthreads();                 // all readers of buf^1 (stage kt-1) done
        if (kt + 1 < ktiles) {
            stage(kt + 1, buf ^ 1);      // overlap DMA of next stage with compute
            wait_asynccnt_le<3>();       // own stage-kt copies landed (3 in flight)
        } else {
            wait_asynccnt_le<0>();
        }
        __syncthreads();                 // every wave's stage-kt copies landed

        const float* as = &As[buf][0];
        const float* bs = &Bs[buf][0];
        // A frag (16x4): lanes 0-15 hold K=ks,ks+1; lanes 16-31 hold K=ks+2,ks+3
        // B frag (4x16): VGPR0 = rows {ks, ks+2}, VGPR1 = rows {ks+1, ks+3}
        #pragma unroll
        for (int kk = 0; kk < KC; kk += 4) {
            const int ks = kk + half * 2;
            v2f a, b;
            a.x = as[(mt * 16 + lm) * LDST + ks];
            a.y = as[(mt * 16 + lm) * LDST + ks + 1];
            if (!TRANS_B) {
                b.x = bs[ ks      * LDST + nt * 16 + lm];
                b.y = bs[(ks + 1) * LDST + nt * 16 + lm];
            } else {   // K-major tile: Bs[n][k]
                b.x = bs[(nt * 16 + lm) * LDST + ks];
                b.y = bs[(nt * 16 + lm) * LDST + ks + 1];
            }
            acc = __builtin_amdgcn_wmma_f32_16x16x4_f32(
                false, a, false, b, (short)0, acc, false, false);
        }
    }

    // ---- store C tile: VGPR r -> M = r + half*8, N = lane&15 ----
    const int col = colBase + nt * 16 + lm;
    #pragma unroll
    for (int r = 0; r < 8; ++r) {
        const int row = rowBase + mt * 16 + half * 8 + r;
        if (row < M) {
            float v = acc[r];
            if (CAUSAL && col > row) v -= 1.0e9f;   // -NEG_INF mask
            Cb[(long)row * ldc + col] = v;
        }
    }
}

// Build X0 = [one_hot(tokens, 512) | eye(1024)] : one block per (b, n) row.
__global__ __launch_bounds__(256)
void build_x0(const int* __restrict__ tok, float* __restrict__ X)
{
    const int rowIdx = blockIdx.x;          // b*CTX + n
    const int n = rowIdx & (CTX - 1);
    const int t = tok[rowIdx];
    float* xr = X + (long)rowIdx * DDIM;
    for (int c = threadIdx.x; c < DDIM; c += 256)
        xr[c] = (c == t || c == VOCAB + n) ? 1.0f : 0.0f;
}

// Row softmax over length-1024 rows; one 8-wave block per row.
__global__ __launch_bounds__(256)
void softmax_rows(float* __restrict__ S)
{
    const long row = blockIdx.x;
    float* s = S + row * (long)CTX;
    const int tid  = threadIdx.x;
    const int wave = tid >> 5;
    const int lane = tid & 31;
    __shared__ float red[8];

    float v[4];
    float m = -3.0e38f;
    #pragma unroll
    for (int i = 0; i < 4; ++i) { v[i] = s[tid + 256 * i]; m = fmaxf(m, v[i]); }
    #pragma unroll
    for (int off = 16; off > 0; off >>= 1) m = fmaxf(m, __shfl_xor(m, off, 32));
    if (lane == 0) red[wave] = m;
    __syncthreads();
    m = red[0];
    #pragma unroll
    for (int w = 1; w < 8; ++w) m = fmaxf(m, red[w]);
    __syncthreads();

    float sum = 0.f;
    #pragma unroll
    for (int i = 0; i < 4; ++i) { v[i] = __expf(v[i] - m); sum += v[i]; }
    #pragma unroll
    for (int off = 16; off > 0; off >>= 1) sum += __shfl_xor(sum, off, 32);
    if (lane == 0) red[wave] = sum;
    __syncthreads();
    float tot = 0.f;
    #pragma unroll
    for (int w = 0; w < 8; ++w) tot += red[w];
    const float inv = 1.0f / tot;
    #pragma unroll
    for (int i = 0; i < 4; ++i) s[tid + 256 * i] = v[i] * inv;
}

extern "C" void kernel_launch(void* const* d_in, const int* in_sizes, int n_in,
                              void* d_out, int out_size, void* d_ws, size_t ws_size,
                              hipStream_t stream)
{
    const int*   tok = (const int*)d_in[0];    // (32, 1024)
    const float* R   = (const float*)d_in[1];  // (4, 1536, 1536)
    const float* U   = (const float*)d_in[2];  // (512, 1536)
    float* out = (float*)d_out;                // (32, 512)

    // Workspace layout (floats): X | E | S  = 201MB + 201MB + 134MB
    float* X = (float*)d_ws;
    float* E = X + (long)BATCH * CTX * DDIM;
    float* S = E + (long)BATCH * CTX * DDIM;

    const dim3 blk(256);
    const long sXD = (long)CTX * DDIM;
    const long sSS = (long)CTX * CTX;

    build_x0<<<dim3(BATCH * CTX), blk, 0, stream>>>(tok, X);

    float* Xc = X;
    float* Xn = E;
    for (int l = 0; l < NLAYERS; ++l) {
        const float* Rl = R + (long)l * DDIM * DDIM;
        // encR = X @ R_l   (into Xn buffer)
        gemm_wmma_f32<false, false><<<dim3(CTX / BM, DDIM / BN, BATCH), blk, 0, stream>>>(
            Xc, Rl, Xn, CTX, DDIM, DDIM, DDIM, DDIM, DDIM, sXD, 0, sXD);
        // S = encR @ X^T + causal mask
        gemm_wmma_f32<true, true><<<dim3(CTX / BM, CTX / BN, BATCH), blk, 0, stream>>>(
            Xn, Xc, S, CTX, CTX, DDIM, DDIM, DDIM, CTX, sXD, sXD, sSS);
        // A = softmax(S)
        softmax_rows<<<dim3(BATCH * CTX), blk, 0, stream>>>(S);
        // X_next = A @ X   (overwrite Xn; encR no longer needed)
        gemm_wmma_f32<false, false><<<dim3(CTX / BM, DDIM / BN, BATCH), blk, 0, stream>>>(
            S, Xc, Xn, CTX, DDIM, CTX, CTX, DDIM, DDIM, sSS, sXD, sXD);
        float* t = Xc; Xc = Xn; Xn = t;
    }

    // logits = X[:, -1, :] @ U^T : M=32 rows of Xc (row stride CTX*DDIM), NT gemm
    gemm_wmma_f32<true, false><<<dim3(1, VOCAB / BN, 1), blk, 0, stream>>>(
        Xc + (long)(CTX - 1) * DDIM, U, out, BATCH, VOCAB, DDIM,
        sXD, DDIM, VOCAB, 0, 0, 0);
}